// MetaGNNNoEdgeAttr_60765197304213
// MI455X (gfx1250) — compile-verified
//
#include <hip/hip_runtime.h>
#include <math.h>

typedef __attribute__((ext_vector_type(2))) float v2f;
typedef __attribute__((ext_vector_type(8))) float v8f;

#define NN     50000
#define EMBD   256
#define NE0    100000
#define NE     250000      // 2*NE0 undirected + NN self loops
#define NTASK  (2*NE)      // edge*head tasks
#define KBIG   256
#define NBIG   768         // [v(256) | kk(2*128) | qq(2*128)]

// ---------------------------------------------------------------------------
// helpers
// ---------------------------------------------------------------------------
__device__ __forceinline__ void edge_sd(const long long* __restrict__ ei, int e, int& s, int& d) {
    if (e < NE0)            { s = (int)ei[e];            d = (int)ei[NE0 + e]; }
    else if (e < 2 * NE0)   { int b = e - NE0; s = (int)ei[NE0 + b]; d = (int)ei[b]; }
    else                    { s = d = e - 2 * NE0; }
}

__device__ __forceinline__ unsigned fkey(float f) {
    unsigned b = __float_as_uint(f);
    return (b & 0x80000000u) ? ~b : (b | 0x80000000u);
}
__device__ __forceinline__ float unkey(unsigned k) {
    unsigned b = (k & 0x80000000u) ? (k & 0x7FFFFFFFu) : ~k;
    return __uint_as_float(b);
}

// packed-B indexing: Bp stores float2 {B[2kp][col], B[2kp+1][col]} at [kp*Ncols + col]
__device__ __forceinline__ size_t packed_idx(int k, int col, int Ncols) {
    return ((size_t)(k >> 1) * Ncols + col) * 2 + (k & 1);
}

// ---------------------------------------------------------------------------
// 0) fold W1 (and 1/sqrt(hd)) into Wkqv -> WbigP (packed float2 pairs), bbig
//    cols [0,256)   : v      = Wkqv[:,512:768]
//    cols [256,512) : kk_h   = (Wkqv[:,256+128h:...]/sqrt(128)) @ W1[0:128]
//    cols [512,768) : qq_h   =  Wkqv[:,128h:...]               @ W1[128:256]  (+b1 in bias)
// ---------------------------------------------------------------------------
__global__ void prep_weights(const float* __restrict__ Wkqv, const float* __restrict__ bkqv,
                             const float* __restrict__ W1,   const float* __restrict__ b1,
                             float* __restrict__ WbigP, float* __restrict__ bbig) {
    int idx = blockIdx.x * blockDim.x + threadIdx.x;
    if (idx >= KBIG * NBIG) return;
    int k = idx / NBIG, col = idx % NBIG;
    const float inv = 0.08838834764831845f;   // 1/sqrt(128)
    float w;
    if (col < 256) {
        w = Wkqv[k * NBIG + 512 + col];
    } else if (col < 512) {
        int h = (col - 256) >> 7, j = (col - 256) & 127;
        const float* wk = Wkqv + k * NBIG + 256 + h * 128;
        float s = 0.f;
        for (int t = 0; t < 128; ++t) s += wk[t] * W1[t * 128 + j];
        w = s * inv;
    } else {
        int h = (col - 512) >> 7, j = (col - 512) & 127;
        const float* wq = Wkqv + k * NBIG + h * 128;
        float s = 0.f;
        for (int t = 0; t < 128; ++t) s += wq[t] * W1[(128 + t) * 128 + j];
        w = s;
    }
    WbigP[packed_idx(k, col, NBIG)] = w;
    if (k == 0) {
        float b;
        if (col < 256) b = bkqv[512 + col];
        else if (col < 512) {
            int h = (col - 256) >> 7, j = (col - 256) & 127;
            float s = 0.f;
            for (int t = 0; t < 128; ++t) s += bkqv[256 + h * 128 + t] * W1[t * 128 + j];
            b = s * inv;
        } else {
            int h = (col - 512) >> 7, j = (col - 512) & 127;
            float s = 0.f;
            for (int t = 0; t < 128; ++t) s += bkqv[h * 128 + t] * W1[(128 + t) * 128 + j];
            b = s + b1[j];
        }
        bbig[col] = b;
    }
}

__global__ void pack_wout(const float* __restrict__ Wout, float* __restrict__ WoutP) {
    int idx = blockIdx.x * blockDim.x + threadIdx.x;
    if (idx >= EMBD * EMBD) return;
    int k = idx / EMBD, col = idx % EMBD;
    WoutP[packed_idx(k, col, EMBD)] = Wout[idx];
}

__global__ void zero_kernel(float* __restrict__ p, long n) {
    long i = (long)blockIdx.x * blockDim.x + threadIdx.x;
    long stride = (long)gridDim.x * blockDim.x;
    for (; i < n; i += stride) p[i] = 0.0f;
}

// ---------------------------------------------------------------------------
// f32 WMMA GEMM: C[M,Ncols] = A[M,K] @ Bp[K,Ncols] (+epilogue)
// Bp pre-packed: float2 {B[k][col], B[k+1][col]} at [(k/2)*Ncols + col]
// each wave -> 16 rows x (16*TILES) cols; per k-step: clause of 1 A-b64 +
// TILES B-b64 loads into distinct regs, then TILES wmma draining the clause.
// mode 0: C = AB + bias[col]
// mode 1: C = relu(AB + deg[row]*bias[col]) + xres[row,col]     (Ncols==256)
// A-frag (16x4 f32): lane%16 = row M; lane half selects K pair {0,1}/{2,3}
// B-frag (4x16 f32): lane%16 = col N; lane half selects K pair
// C/D: VGPR g -> M = g + (lane>=16 ? 8 : 0), N = lane%16
// ---------------------------------------------------------------------------
template <int TILES>
__global__ void gemm_f32_wmma(const float* __restrict__ A, const v2f* __restrict__ Bp,
                              const float* __restrict__ bias, float* __restrict__ C,
                              int K, int Ncols, int mode,
                              const float* __restrict__ deg, const float* __restrict__ xres) {
    const int lane  = threadIdx.x;
    const int mbase = blockIdx.x * 16;
    const int nbase = (blockIdx.y * blockDim.y + threadIdx.y) * (16 * TILES);
    const int l16   = lane & 15;
    const int koff  = (lane >> 4) << 1;           // 0 or 2

    const float* __restrict__ arow = A + (size_t)(mbase + l16) * K + koff;

    v8f zero8 = {0.f,0.f,0.f,0.f,0.f,0.f,0.f,0.f};
    v8f acc[TILES];
#pragma unroll
    for (int j = 0; j < TILES; ++j) acc[j] = zero8;

    for (int k = 0; k < K; k += 4) {
        // stage all fragments first -> distinct registers, clause-able loads
        const v2f* __restrict__ bp = Bp + (size_t)((k + koff) >> 1) * Ncols + nbase + l16;
        v2f a;
        a.x = arow[k];
        a.y = arow[k + 1];
        v2f b[TILES];
#pragma unroll
        for (int j = 0; j < TILES; ++j) b[j] = bp[j * 16];
#pragma unroll
        for (int j = 0; j < TILES; ++j) {
            acc[j] = __builtin_amdgcn_wmma_f32_16x16x4_f32(
                false, a, false, b[j], (short)0, acc[j], false, false);
        }
    }

    const int rowhi = (lane >> 4) << 3;           // 0 or 8
#pragma unroll
    for (int j = 0; j < TILES; ++j) {
#pragma unroll
        for (int g = 0; g < 8; ++g) {
            int row = mbase + rowhi + g;
            int col = nbase + j * 16 + l16;
            float v = acc[j][g];
            if (mode == 0) {
                v += bias[col];
            } else {
                v += deg[row] * bias[col];
                v = fmaxf(v, 0.0f) + xres[(size_t)row * EMBD + col];
            }
            C[(size_t)row * Ncols + col] = v;
        }
    }
}

// ---------------------------------------------------------------------------
// per (edge,head): a = relu(kk[src,h] + qq[dst,h]) . W2 + b2 ; segment max
// one wave per task, float4 per lane
// ---------------------------------------------------------------------------
__global__ void edge_logits(const long long* __restrict__ ei, const float* __restrict__ C1,
                            const float* __restrict__ W2, const float* __restrict__ b2,
                            float* __restrict__ Alog, unsigned* __restrict__ mkey) {
    int task = blockIdx.x * blockDim.y + threadIdx.y;
    if (task >= NTASK) return;
    int e = task >> 1, h = task & 1;
    int s, d; edge_sd(ei, e, s, d);
    int lane = threadIdx.x;

    const float4 kv = ((const float4*)(C1 + (size_t)s * NBIG + 256 + h * 128))[lane];
    const float4 qv = ((const float4*)(C1 + (size_t)d * NBIG + 512 + h * 128))[lane];
    const float4 w  = ((const float4*)W2)[lane];

    float p = fmaxf(kv.x + qv.x, 0.f) * w.x + fmaxf(kv.y + qv.y, 0.f) * w.y +
              fmaxf(kv.z + qv.z, 0.f) * w.z + fmaxf(kv.w + qv.w, 0.f) * w.w;
#pragma unroll
    for (int off = 16; off > 0; off >>= 1) p += __shfl_down(p, off, 32);

    if (lane == 0) {
        float a = p + b2[0];
        Alog[task] = a;
        atomicMax(&mkey[d * 2 + h], fkey(a));
    }
}

// ---------------------------------------------------------------------------
// ex = exp(a - max) ; segment sum
// ---------------------------------------------------------------------------
__global__ void edge_exp(const long long* __restrict__ ei, const float* __restrict__ Alog,
                         const unsigned* __restrict__ mkey,
                         float* __restrict__ EX, float* __restrict__ ssum) {
    int t = blockIdx.x * blockDim.x + threadIdx.x;
    if (t >= NTASK) return;
    int e = t >> 1, h = t & 1;
    int s, d; edge_sd(ei, e, s, d);
    float m  = unkey(mkey[d * 2 + h]);
    float ex = expf(Alog[t] - m);
    EX[t] = ex;
    atomicAdd(&ssum[d * 2 + h], ex);
}

// ---------------------------------------------------------------------------
// agg[dst] += alpha_h * v[src]  (wave per edge, both heads), deg[dst] += 1
// ---------------------------------------------------------------------------
__global__ void edge_agg(const long long* __restrict__ ei, const float* __restrict__ C1,
                         const float* __restrict__ EX, const float* __restrict__ ssum,
                         float* __restrict__ agg, float* __restrict__ deg) {
    int e = blockIdx.x * blockDim.y + threadIdx.y;
    if (e >= NE) return;
    int s, d; edge_sd(ei, e, s, d);
    int lane = threadIdx.x;

    float a0 = EX[e * 2 + 0] / (ssum[d * 2 + 0] + 1e-16f);
    float a1 = EX[e * 2 + 1] / (ssum[d * 2 + 1] + 1e-16f);

    const float* vp = C1 + (size_t)s * NBIG;           // v row (cols 0..255)
    float*       ap = agg + (size_t)d * EMBD;
    int o = lane * 4;
    float4 v0 = *(const float4*)(vp + o);
    float4 v1 = *(const float4*)(vp + 128 + o);

    atomicAdd(ap + o + 0, a0 * v0.x);
    atomicAdd(ap + o + 1, a0 * v0.y);
    atomicAdd(ap + o + 2, a0 * v0.z);
    atomicAdd(ap + o + 3, a0 * v0.w);
    atomicAdd(ap + 128 + o + 0, a1 * v1.x);
    atomicAdd(ap + 128 + o + 1, a1 * v1.y);
    atomicAdd(ap + 128 + o + 2, a1 * v1.z);
    atomicAdd(ap + 128 + o + 3, a1 * v1.w);
    if (lane == 0) atomicAdd(&deg[d], 1.0f);
}

// ---------------------------------------------------------------------------
extern "C" void kernel_launch(void* const* d_in, const int* in_sizes, int n_in,
                              void* d_out, int out_size, void* d_ws, size_t ws_size,
                              hipStream_t stream) {
    const float*     x    = (const float*)d_in[0];
    const long long* ei   = (const long long*)d_in[1];
    const float*     Wkqv = (const float*)d_in[2];
    const float*     bkqv = (const float*)d_in[3];
    const float*     W1   = (const float*)d_in[4];
    const float*     b1   = (const float*)d_in[5];
    const float*     W2   = (const float*)d_in[6];
    const float*     b2   = (const float*)d_in[7];
    const float*     Wout = (const float*)d_in[8];
    const float*     bout = (const float*)d_in[9];
    float*           out  = (float*)d_out;

    // workspace partition (floats)
    float*    ws    = (float*)d_ws;
    float*    WbigP = ws;                                   // 256*768 (packed pairs)
    float*    bbig  = WbigP + (size_t)KBIG * NBIG;          // 768
    float*    WoutP = bbig + NBIG;                          // 256*256 (packed pairs)
    float*    C1    = WoutP + (size_t)EMBD * EMBD;          // 50000*768
    float*    Alog  = C1 + (size_t)NN * NBIG;               // 500000
    float*    EX    = Alog + NTASK;                         // 500000
    unsigned* mkey  = (unsigned*)(EX + NTASK);              // 100000
    float*    ssum  = (float*)(mkey + 2 * NN);              // 100000
    float*    deg   = ssum + 2 * NN;                        // 50000
    float*    agg   = deg + NN;                             // 50000*256

    // 0) fold W1 into Wkqv (packed), pack Wout
    prep_weights<<<(KBIG * NBIG + 255) / 256, 256, 0, stream>>>(Wkqv, bkqv, W1, b1, WbigP, bbig);
    pack_wout<<<(EMBD * EMBD + 255) / 256, 256, 0, stream>>>(Wout, WoutP);

    // 1) zero mkey|ssum|deg|agg (contiguous)
    long nz = 2L * NN + 2L * NN + NN + (long)NN * EMBD;
    zero_kernel<<<2048, 256, 0, stream>>>((float*)mkey, nz);

    // 2) C1[N,768] = x @ Wbig + bbig   (f32 WMMA, 6 waves x 128 cols)
    {
        dim3 grid(NN / 16, 1);            // 3125 x 1
        dim3 block(32, 6);                // 6 waves * 128 cols = 768
        gemm_f32_wmma<8><<<grid, block, 0, stream>>>(x, (const v2f*)WbigP, bbig, C1,
                                                     KBIG, NBIG, 0, nullptr, nullptr);
    }

    // 3) logits + segment max
    edge_logits<<<NTASK / 8, dim3(32, 8), 0, stream>>>(ei, C1, W2, b2, Alog, mkey);

    // 4) exp + segment sum
    edge_exp<<<(NTASK + 255) / 256, 256, 0, stream>>>(ei, Alog, mkey, EX, ssum);

    // 5) weighted aggregation + degree
    edge_agg<<<NE / 8, dim3(32, 8), 0, stream>>>(ei, C1, EX, ssum, agg, deg);

    // 6) out = relu(agg @ Wout + deg*bout) + x   (f32 WMMA, 2 waves x 128 cols)
    {
        dim3 grid(NN / 16, 1);
        dim3 block(32, 2);                // 2 waves * 128 cols = 256
        gemm_f32_wmma<8><<<grid, block, 0, stream>>>(agg, (const v2f*)WoutP, bout, out,
                                                     EMBD, EMBD, 1, deg, x);
    }
}